// Upsampling_1125281432060
// MI455X (gfx1250) — compile-verified
//
#include <hip/hip_runtime.h>

#define RDIM 128
#define CH 32

typedef float v2f __attribute__((ext_vector_type(2)));
typedef float v8f __attribute__((ext_vector_type(8)));

typedef __attribute__((address_space(1))) int* as1i;
typedef __attribute__((address_space(3))) int* as3i;

__global__ void fill_table_kernel(int* table, int n) {
    int i = blockIdx.x * 256 + threadIdx.x;
    if (i < n) table[i] = -1;
}

__global__ void build_table_kernel(const int* __restrict__ cc, int* __restrict__ table, int n) {
    int i = blockIdx.x * 256 + threadIdx.x;
    if (i < n) {
        int x = cc[i * 3 + 0], y = cc[i * 3 + 1], z = cc[i * 3 + 2];
        table[(x * RDIM + y) * RDIM + z] = i;
    }
}

// One wave32 per coarse parent: out[8 children x 32ch] = W[8x27] x F[27x32]
// via V_WMMA_F32_16X16X4_F32 chains. Feature tile staged TRANSPOSED (Ft[chan][k])
// directly into LDS with global_load_async_to_lds_b32, so every WMMA operand is a
// single contiguous ds_load_b64.
__global__ __launch_bounds__(256) void upsample_wmma_kernel(
    const float* __restrict__ feat, const int* __restrict__ cc,
    const int* __restrict__ fc, const int* __restrict__ table,
    float* __restrict__ out, int n_parents) {
    __shared__ float sFt[8][32][28];  // per-wave: 32 channels x 28 K-slots (27 + zero pad)
    __shared__ float sW[8][16][28];   // per-wave: A matrix, rows 8..15 stay zero

    const int wave = threadIdx.x >> 5;
    const int lane = threadIdx.x & 31;
    const int p = blockIdx.x * 8 + wave;
    if (p >= n_parents) return;       // wave-uniform exit (EXEC stays all-ones)

    float (*Ft)[28] = sFt[wave];
    float (*W)[28]  = sW[wave];

    const int cx = cc[p * 3 + 0], cy = cc[p * 3 + 1], cz = cc[p * 3 + 2];

    // lanes 0..26: hash-table lookup for the 27-neighborhood
    int myIdx = -1;
    if (lane < 27) {
        int dx = lane / 9 - 1, dy = (lane / 3) % 3 - 1, dz = lane % 3 - 1;
        int nx = cx + dx, ny = cy + dy, nz = cz + dz;
        if ((unsigned)nx < RDIM && (unsigned)ny < RDIM && (unsigned)nz < RDIM)
            myIdx = table[(nx * RDIM + ny) * RDIM + nz];
    }

    // Stage feature rows (transposed) into LDS; zero column == masked/missing neighbor.
    // Global side stays coalesced (128B row); LDS side is the per-lane transpose scatter.
    for (int k = 0; k < 28; ++k) {
        int row = (k < 27) ? __shfl(myIdx, k, 32) : -1;
        if (row >= 0) {
#if __has_builtin(__builtin_amdgcn_global_load_async_to_lds_b32)
            __builtin_amdgcn_global_load_async_to_lds_b32(
                (as1i)(void*)(feat + (long)row * CH + lane),
                (as3i)(void*)(&Ft[lane][k]), 0, 0);
#else
            Ft[lane][k] = feat[(long)row * CH + lane];
#endif
        } else {
            Ft[lane][k] = 0.0f;
        }
    }

    // Zero the A (weight) matrix
    float* wflat = &W[0][0];
    for (int i = lane; i < 16 * 28; i += 32) wflat[i] = 0.0f;
    __builtin_amdgcn_wave_barrier();

    // Trilinear weights, computed generically from fine_coords: 64 (child,corner) pairs
    for (int q = lane; q < 64; q += 32) {
        int child = q >> 3, corner = q & 7;
        long m = (long)p * 8 + child;
        float w = 1.0f; int kk = 0; bool ok = true;
        #pragma unroll
        for (int a = 0; a < 3; ++a) {
            int fcoord = fc[m * 3 + a];
            float g = ((float)fcoord + 0.5f) * 0.5f - 0.5f;  // VS_FINE=0.5, VS_COARSE=1.0
            float b = floorf(g);
            float fr = g - b;
            int bit = (corner >> (2 - a)) & 1;
            w *= bit ? fr : (1.0f - fr);
            int ca = (a == 0) ? cx : ((a == 1) ? cy : cz);
            int d = (int)b + bit - ca + 1;                   // neighbor delta in [0,2]
            if ((unsigned)d > 2u) ok = false;
            kk = kk * 3 + d;
        }
        if (ok) W[child][kk] = w;                            // distinct kk per corner: no races
    }
    __builtin_amdgcn_wave_barrier();
    asm volatile("s_wait_asynccnt 0x0" ::: "memory");
    asm volatile("s_wait_dscnt 0x0" ::: "memory");

    // WMMA f32 16x16x4: lanes<16 hold K=k0,k0+1 ; lanes>=16 hold K=k0+2,k0+3.
    // Both A and B fragments are contiguous, 8B-aligned LDS pairs -> ds_load_b64.
    const int r16 = lane & 15;
    const int kh  = (lane >> 4) << 1;
    v8f acc0 = {0.f, 0.f, 0.f, 0.f, 0.f, 0.f, 0.f, 0.f};
    v8f acc1 = acc0;
    #pragma unroll
    for (int k0 = 0; k0 < 28; k0 += 4) {
        int kq = k0 + kh;
        v2f a  = *(const v2f*)&W[r16][kq];
        v2f b0 = *(const v2f*)&Ft[r16][kq];
        v2f b1 = *(const v2f*)&Ft[16 + r16][kq];
        acc0 = __builtin_amdgcn_wmma_f32_16x16x4_f32(false, a, false, b0, (short)0, acc0, false, false);
        acc1 = __builtin_amdgcn_wmma_f32_16x16x4_f32(false, a, false, b1, (short)0, acc1, false, false);
    }

    // Repack so the full wave issues coalesced 128B row stores (kernel is store-bound):
    // lanes 0..15 carry chans 0..15 (acc0), lanes 16..31 take chans 16..31 from lane-16's acc1.
    #pragma unroll
    for (int j = 0; j < 8; ++j) {
        float hi = __shfl_up(acc1[j], 16, 32);
        float v  = (lane < 16) ? acc0[j] : hi;
        out[((long)p * 8 + j) * CH + lane] = v;
    }
}

extern "C" void kernel_launch(void* const* d_in, const int* in_sizes, int n_in,
                              void* d_out, int out_size, void* d_ws, size_t ws_size,
                              hipStream_t stream) {
    const float* feat = (const float*)d_in[0];
    const int*   cc   = (const int*)d_in[1];
    const int*   fc   = (const int*)d_in[2];
    float*       out  = (float*)d_out;

    const int n_parents = in_sizes[1] / 3;          // N_IN
    int* table = (int*)d_ws;                        // 128^3 * 4B = 8 MiB scratch
    const int tsize = RDIM * RDIM * RDIM;

    fill_table_kernel<<<(tsize + 255) / 256, 256, 0, stream>>>(table, tsize);
    build_table_kernel<<<(n_parents + 255) / 256, 256, 0, stream>>>(cc, table, n_parents);
    upsample_wmma_kernel<<<(n_parents + 7) / 8, 256, 0, stream>>>(feat, cc, fc, table, out, n_parents);
}